// RefineModule2_68496138436771
// MI455X (gfx1250) — compile-verified
//
#include <hip/hip_runtime.h>
#include <hip/hip_bf16.h>

#define BATCH 4
#define FC    3
#define MC    21
#define IMH   128
#define IMW   128
#define KSIZE 7
#define HPAD  3
#define NB    48
#define NITER 10
#define TILE  16
#define HALO  (TILE + KSIZE - 1)   // 22
#define PWD   136                  // padded row stride (128 + 6 + 2 slack)
#define PHT   134                  // padded height (128 + 6)
#define PLANE (PHT * PWD)

#define AFF_ELEMS ((size_t)BATCH * NB * IMH * IMW)   // 3,145,728 floats
#define PF_ELEMS  ((size_t)BATCH * FC * PLANE)       //   218,688 floats
#define PAD_ELEMS ((size_t)BATCH * MC * PLANE)       // 1,530,816 floats

#ifndef __has_builtin
#define __has_builtin(x) 0
#endif
#if defined(__HIP_DEVICE_COMPILE__) && defined(__AMDGCN__) &&      \
    __has_builtin(__builtin_amdgcn_tensor_load_to_lds) &&          \
    __has_builtin(__builtin_amdgcn_s_wait_tensorcnt)
#define USE_TDM 1
#else
#define USE_TDM 0
#endif

// Proof-of-path: fail the gfx1250 device pass if the TDM builtins are absent.
// (Host pass never sees this: __HIP_DEVICE_COMPILE__ is device-only.)
#if defined(__HIP_DEVICE_COMPILE__) && defined(__AMDGCN__) && !USE_TDM
#error "CDNA5 TDM builtins not available on gfx1250 device pass"
#endif

__device__ __forceinline__ int iclampi(int v, int lo, int hi) {
  return v < lo ? lo : (v > hi ? hi : v);
}

// ---------------------------------------------------------------------------
// TDM issue: 2D tile (22 rows x 22 cols of f32, src row stride PWD) -> LDS.
// Descriptor bitfields per CDNA5 ISA ch.8 (D# groups 0/1); groups 2/3 zero
// (tile_dim2 == 0 -> 2D walk). Tiles are always fully interior of the padded
// buffers, so OOB never triggers.
// ---------------------------------------------------------------------------
#if USE_TDM
typedef unsigned int tdm_v4u __attribute__((ext_vector_type(4)));
typedef int          tdm_v4i __attribute__((ext_vector_type(4)));
typedef int          tdm_v8i __attribute__((ext_vector_type(8)));

#define TDM_TDIM0 4096u
#define TDM_TDIM1 4096u

__device__ __forceinline__ void tdm_issue(const float* gsrc, unsigned lds_addr) {
  unsigned long long ga = (unsigned long long)(size_t)gsrc;
  tdm_v4u g0;
  g0.x = 1u;                                         // count = 1 (valid D#)
  g0.y = lds_addr;                                   // lds_addr (bytes)
  g0.z = (unsigned)(ga & 0xFFFFFFFFull);             // global_addr[31:0]
  g0.w = (unsigned)((ga >> 32) & 0x01FFFFFFull)      // global_addr[56:32]
         | (2u << 30);                               // type = 2 ("image")
  tdm_v8i g1;
  g1[0] = (int)(2u << 16);                           // data_size = 2 (4 bytes)
  g1[1] = (int)((TDM_TDIM0 & 0xFFFFu) << 16);        // tensor_dim0[15:0]
  g1[2] = (int)(((TDM_TDIM0 >> 16) & 0xFFFFu) |      // tensor_dim0[31:16]
                ((TDM_TDIM1 & 0xFFFFu) << 16));      // tensor_dim1[15:0]
  g1[3] = (int)(((TDM_TDIM1 >> 16) & 0xFFFFu) |      // tensor_dim1[31:16]
                ((unsigned)HALO << 16));             // tile_dim0 = 22
  g1[4] = (int)HALO;                                 // tile_dim1 = 22, tile_dim2 = 0
  g1[5] = (int)PWD;                                  // tensor_dim0_stride = 136
  g1[6] = 0;
  g1[7] = 0;
  tdm_v4i gz = {0, 0, 0, 0};
#if defined(__clang_major__) && (__clang_major__ >= 23)
  tdm_v8i gz8 = {0, 0, 0, 0, 0, 0, 0, 0};
  __builtin_amdgcn_tensor_load_to_lds(g0, g1, gz, gz, gz8, 0);
#else
  __builtin_amdgcn_tensor_load_to_lds(g0, g1, gz, gz, 0);
#endif
}
#endif // USE_TDM

// ---------------------------------------------------------------------------
// 48-tap weighted window dot product from LDS halo tile (7x7 minus center).
// ---------------------------------------------------------------------------
__device__ __forceinline__ float win_dot(const float* __restrict__ T,
                                         const float (&w)[NB], int ty, int tx) {
  float acc = 0.f;
  int n = 0;
#pragma unroll
  for (int dy = 0; dy < KSIZE; ++dy) {
#pragma unroll
    for (int dx = 0; dx < KSIZE; ++dx) {
      if (dy == HPAD && dx == HPAD) continue;
      acc = fmaf(w[n], T[(ty + dy) * HALO + (tx + dx)], acc);
      ++n;
    }
  }
  return acc;
}

// ---------------------------------------------------------------------------
// Pad a 128x128 plane into a PHTxPWD edge-replicated plane (feats and mask).
// ---------------------------------------------------------------------------
__global__ __launch_bounds__(256) void pad_fill_kernel(const float* __restrict__ m,
                                                       float* __restrict__ P) {
  const int plane = blockIdx.x;
  const float* src = m + (size_t)plane * IMH * IMW;
  float* dst = P + (size_t)plane * PLANE;
  for (int i = threadIdx.x; i < PLANE; i += blockDim.x) {
    int y = i / PWD, x = i - y * PWD;
    int sy = iclampi(y - HPAD, 0, IMH - 1);
    int sx = iclampi(x - HPAD, 0, IMW - 1);
    dst[i] = src[sy * IMW + sx];
  }
}

// ---------------------------------------------------------------------------
// Kernel 1: per-pixel 48-way softmax affinity from padded feats (TDM tiles).
// ---------------------------------------------------------------------------
__global__ __launch_bounds__(256) void aff_kernel(const float* __restrict__ pf,
                                                  float* __restrict__ aff) {
  __shared__ __align__(16) float ft[FC][HALO * HALO];
  const int tx = threadIdx.x, ty = threadIdx.y;
  const int tid = ty * TILE + tx;
  const int b = blockIdx.z;
  const int X0 = blockIdx.x * TILE, Y0 = blockIdx.y * TILE;

#if USE_TDM
  if (tid < 32) {                       // wave 0 drives the Tensor Data Mover
#pragma unroll
    for (int c = 0; c < FC; ++c)
      tdm_issue(pf + (size_t)(b * FC + c) * PLANE + Y0 * PWD + X0,
                (unsigned)(size_t)&ft[c][0]);
    __builtin_amdgcn_s_wait_tensorcnt(0);
  }
  __syncthreads();
#else
  for (int c = 0; c < FC; ++c) {
    const float* src = pf + (size_t)(b * FC + c) * PLANE + Y0 * PWD + X0;
    for (int i = tid; i < HALO * HALO; i += TILE * TILE) {
      int ly = i / HALO, lx = i - ly * HALO;
      ft[c][i] = src[ly * PWD + lx];
    }
  }
  __syncthreads();
#endif

  const int cbase = (ty + HPAD) * HALO + (tx + HPAD);
  const float q0 = ft[0][cbase], q1 = ft[1][cbase], q2 = ft[2][cbase];

  // Per-channel std over the 48 neighbors (ddof=1).
  float s0 = 0.f, t0 = 0.f, s1 = 0.f, t1 = 0.f, s2 = 0.f, t2 = 0.f;
#pragma unroll
  for (int dy = 0; dy < KSIZE; ++dy) {
#pragma unroll
    for (int dx = 0; dx < KSIZE; ++dx) {
      if (dy == HPAD && dx == HPAD) continue;
      int o = (ty + dy) * HALO + (tx + dx);
      float k0 = ft[0][o], k1 = ft[1][o], k2 = ft[2][o];
      s0 += k0; t0 += k0 * k0;
      s1 += k1; t1 += k1 * k1;
      s2 += k2; t2 += k2 * k2;
    }
  }
  const float invN = 1.f / (float)NB, invN1 = 1.f / (float)(NB - 1);
  float v0 = (t0 - s0 * s0 * invN) * invN1; v0 = v0 > 0.f ? v0 : 0.f;
  float v1 = (t1 - s1 * s1 * invN) * invN1; v1 = v1 > 0.f ? v1 : 0.f;
  float v2 = (t2 - s2 * s2 * invN) * invN1; v2 = v2 > 0.f ? v2 : 0.f;
  const float w0 = 1.f / (1e-8f + 0.1f * sqrtf(v0));
  const float w1 = 1.f / (1e-8f + 0.1f * sqrtf(v1));
  const float w2 = 1.f / (1e-8f + 0.1f * sqrtf(v2));

  // Pre-softmax logits: mean over channels of -|k-q|/denom.
  float a[NB];
  float mx = -3.402823e38f;
  {
    int n = 0;
#pragma unroll
    for (int dy = 0; dy < KSIZE; ++dy) {
#pragma unroll
      for (int dx = 0; dx < KSIZE; ++dx) {
        if (dy == HPAD && dx == HPAD) continue;
        int o = (ty + dy) * HALO + (tx + dx);
        float d = fabsf(ft[0][o] - q0) * w0 + fabsf(ft[1][o] - q1) * w1 +
                  fabsf(ft[2][o] - q2) * w2;
        float val = d * (-1.f / 3.f);
        a[n] = val;
        mx = fmaxf(mx, val);
        ++n;
      }
    }
  }
  float tot = 0.f;
#pragma unroll
  for (int n = 0; n < NB; ++n) { a[n] = __expf(a[n] - mx); tot += a[n]; }
  const float inv = 1.f / tot;

  const int gy = Y0 + ty, gx = X0 + tx;
#pragma unroll
  for (int n = 0; n < NB; ++n)
    aff[((b * NB + n) * IMH + gy) * IMW + gx] = a[n] * inv;   // coalesced per n
}

// ---------------------------------------------------------------------------
// Kernel 2: one propagation iteration.
// Per block: 16x16 pixels; 48 affinity weights per thread live in VGPRs for
// all 21 channels. Per-channel 22x22 halo tiles stream through a 3-deep LDS
// pipeline fed by the Tensor Data Mover (lookahead 2, one barrier/channel):
//   - s_wait_tensorcnt(1) before the barrier retires DMA for channel c
//     (TDM completes in-order per wave);
//   - the barrier guarantees all readers of channel c-1 finished before the
//     DMA into buf[(c+2)%3] (same physical buffer) is issued.
// Non-final iterations also replicate boundary outputs into the halo ring so
// no separate padding kernel is needed; the final iteration writes d_out.
// ---------------------------------------------------------------------------
__global__ __launch_bounds__(256) void iter_kernel(const float* __restrict__ Pin,
                                                   const float* __restrict__ aff,
                                                   float* __restrict__ Pout,
                                                   float* __restrict__ outF,
                                                   int final_flag) {
  __shared__ __align__(16) float tile[3][HALO * HALO];
  const int tx = threadIdx.x, ty = threadIdx.y;
  const int tid = ty * TILE + tx;
  const int b = blockIdx.z;
  const int X0 = blockIdx.x * TILE, Y0 = blockIdx.y * TILE;
  const int gx = X0 + tx, gy = Y0 + ty;

  float w[NB];
#pragma unroll
  for (int n = 0; n < NB; ++n)
    w[n] = aff[((b * NB + n) * IMH + gy) * IMW + gx];

  const float* bin = Pin + (size_t)b * MC * PLANE + Y0 * PWD + X0; // halo origin
  float* bout = Pout + (size_t)b * MC * PLANE;

#if USE_TDM
  const bool leader = (tid < 32);
  if (leader) {
    tdm_issue(bin, (unsigned)(size_t)&tile[0][0]);
    tdm_issue(bin + PLANE, (unsigned)(size_t)&tile[1][0]);
  }
  for (int c = 0; c < MC; ++c) {
    if (leader) {
      if (c < MC - 1) __builtin_amdgcn_s_wait_tensorcnt(1);  // channel c landed
      else            __builtin_amdgcn_s_wait_tensorcnt(0);
    }
    __syncthreads();            // publish tile c; readers of c-1 all done
    if (leader && c + 2 < MC)
      tdm_issue(bin + (size_t)(c + 2) * PLANE,
                (unsigned)(size_t)&tile[(c + 2) % 3][0]);
    float acc = win_dot(tile[c % 3], w, ty, tx);
#else
  for (int c = 0; c < MC; ++c) {
    const float* src = bin + (size_t)c * PLANE;
    for (int i = tid; i < HALO * HALO; i += TILE * TILE) {
      int ly = i / HALO, lx = i - ly * HALO;
      tile[0][i] = src[ly * PWD + lx];
    }
    __syncthreads();
    float acc = win_dot(tile[0], w, ty, tx);
#endif
    if (final_flag) {
      outF[((size_t)(b * MC + c) * IMH + gy) * IMW + gx] = acc;
    } else {
      // Interior store + fused edge-replication into the halo ring.
      float* dstc = bout + (size_t)c * PLANE;
      const int py0 = gy + HPAD, px0 = gx + HPAD;
      const int ylo = (gy == 0) ? 0 : py0;
      const int yhi = (gy == IMH - 1) ? (PHT - 1) : py0;
      const int xlo = (gx == 0) ? 0 : px0;
      const int xhi = (gx == IMW - 1) ? (HPAD + IMW + HPAD - 1) : px0;
      for (int py = ylo; py <= yhi; ++py)
        for (int px = xlo; px <= xhi; ++px)
          dstc[py * PWD + px] = acc;
    }
#if !USE_TDM
    __syncthreads();            // fallback: single buffer, drain before refill
#endif
  }
}

// ---------------------------------------------------------------------------
extern "C" void kernel_launch(void* const* d_in, const int* in_sizes, int n_in,
                              void* d_out, int out_size, void* d_ws, size_t ws_size,
                              hipStream_t stream) {
  (void)in_sizes; (void)n_in; (void)out_size; (void)ws_size;
  const float* feats = (const float*)d_in[0];   // (4,3,128,128) f32
  const float* mask  = (const float*)d_in[1];   // (4,21,128,128) f32
  float* out = (float*)d_out;                   // (4,21,128,128) f32

  float* aff = (float*)d_ws;                    // 12.58 MB
  float* PF  = aff + AFF_ELEMS;                 //  0.87 MB (padded feats)
  float* P0  = PF + PF_ELEMS;                   //  6.12 MB (padded ping)
  float* P1  = P0 + PAD_ELEMS;                  //  6.12 MB (padded pong)

  dim3 blk(TILE, TILE);
  dim3 grd(IMW / TILE, IMH / TILE, BATCH);

  pad_fill_kernel<<<dim3(BATCH * FC), 256, 0, stream>>>(feats, PF);
  pad_fill_kernel<<<dim3(BATCH * MC), 256, 0, stream>>>(mask, P0);
  aff_kernel<<<grd, blk, 0, stream>>>(PF, aff);

  float* cur = P0;
  float* nxt = P1;
  for (int t = 0; t < NITER; ++t) {
    const int fin = (t == NITER - 1) ? 1 : 0;
    iter_kernel<<<grd, blk, 0, stream>>>(cur, aff, nxt, out, fin);
    float* tmp = cur; cur = nxt; nxt = tmp;
  }
}